// JointMembership_FuzzyLayer_89816356094196
// MI455X (gfx1250) — compile-verified
//
#include <hip/hip_runtime.h>

// ---- problem constants (from reference) ----
#define BATCH   4096
#define PAIRS   460
#define CLASSES 10
#define PIXDIM  3072
#define TOTAL_PAIRS (BATCH * PAIRS)          // 1,884,160
#define WAVES_PER_BLOCK 8
#define PAIRS_PER_WAVE  16
#define PAIRS_PER_BLOCK (WAVES_PER_BLOCK * PAIRS_PER_WAVE)  // 128 -> 14720 blocks, exact
#define OUT_PER_PAIR (2 * CLASSES)           // 20 contiguous floats per pair in output

typedef __attribute__((ext_vector_type(2))) float v2f;
typedef __attribute__((ext_vector_type(8))) float v8f;

// q[b, 2p,   c] = 0.5*(cos(x0)*cos(th_c) - sin(x0)*sin(x1)*sin(th_c)) + 0.5
// q[b, 2p+1, c] = 0.5*(cos(x0)*cos(x1)) + 0.5            (theta-independent)
__global__ __launch_bounds__(256) void fuzzy_qlayer_wmma(
    const float* __restrict__ x,        // [B,3,32,32] flat
    const int*   __restrict__ pair_idx, // [B,P,2] flat
    const float* __restrict__ theta,    // [C]
    float*       __restrict__ out)      // [B,2P,C] flat == [pair][20]
{
    __shared__ float lds[WAVES_PER_BLOCK * PAIRS_PER_WAVE * OUT_PER_PAIR]; // 8*320 f32

    const int tid    = threadIdx.x;
    const int wave   = tid >> 5;
    const int lane   = tid & 31;
    const int hf     = lane >> 4;    // 0: lanes 0-15 (K=0,1)  1: lanes 16-31 (K=2,3)
    const int lanelo = lane & 15;

    const int g0     = (blockIdx.x * WAVES_PER_BLOCK + wave) * PAIRS_PER_WAVE;
    const int pairId = g0 + lanelo;  // both wave halves mirror the same 16 pairs

    // ---- gather: two pixels of this pair (indices local to one 12KB image row) ----
    const int   b  = pairId / PAIRS;
    const int   i0 = pair_idx[pairId * 2 + 0];
    const int   i1 = pair_idx[pairId * 2 + 1];
    const float x0 = x[b * PIXDIM + i0];
    const float x1 = x[b * PIXDIM + i1];

    const float cx0 = __cosf(x0);
    const float cx1 = __cosf(x1);
    const float sx0 = __sinf(x0);
    const float sx1 = __sinf(x1);

    // ---- A matrix 16x4 f32 (ISA layout: lanes0-15 K=0,1 ; lanes16-31 K=2,3) ----
    // A row M = [cos x0, sin x0*sin x1, (unused), 1.0]
    v2f A;
    if (hf == 0) { A.x = cx0;       A.y = sx0 * sx1; }
    else         { A.x = 0.0f;      A.y = 1.0f;      }

    // ---- B matrix 4x16 f32 from theta (mirrored layout assumption) ----
    // rows: K0 = 0.5*cos(th_c), K1 = -0.5*sin(th_c), K2 = 0, K3 = 0.5  (cols c<10, else 0)
    const float tn = (lanelo < CLASSES) ? theta[lanelo] : 0.0f;
    v2f Bm;
    if (hf == 0) {
        Bm.x = (lanelo < CLASSES) ?  0.5f * __cosf(tn) : 0.0f;  // K=0 row
        Bm.y = (lanelo < CLASSES) ? -0.5f * __sinf(tn) : 0.0f;  // K=1 row
    } else {
        Bm.x = 0.0f;                                            // K=2 row
        Bm.y = (lanelo < CLASSES) ?  0.5f : 0.0f;               // K=3 row (+0.5 bias)
    }

    // EXEC is all-1s here (exact grid, no divergence yet) as WMMA requires.
    v8f acc = {};
    acc = __builtin_amdgcn_wmma_f32_16x16x4_f32(
        /*neg_a=*/false, A, /*neg_b=*/false, Bm,
        /*c_mod=*/(short)0, acc, /*reuse_a=*/false, /*reuse_b=*/false);

    // ---- stage the 16x20 output tile in LDS (row = pair, 20 floats each) ----
    float* wls = &lds[wave * (PAIRS_PER_WAVE * OUT_PER_PAIR)];

    // q0: D[M][c] lives in VGPR r = M%8, lanes (M<8 ? 0-15 : 16-31), N = lanelo
    if (lanelo < CLASSES) {
        #pragma unroll
        for (int r = 0; r < 8; ++r) {
            const int M = r + hf * 8;
            wls[M * OUT_PER_PAIR + lanelo] = acc[r];
        }
    }
    // q1: theta-independent, one value per pair, lanes 0-15 broadcast over 10 classes
    if (hf == 0) {
        const float q1 = 0.5f * cx0 * cx1 + 0.5f;
        const v2f q = { q1, q1 };
        #pragma unroll
        for (int c = 0; c < CLASSES; c += 2)
            *(v2f*)&wls[lanelo * OUT_PER_PAIR + CLASSES + c] = q;
    }

    __syncthreads();  // order LDS stage vs. coalesced flush (also cross-lane safety)

    // ---- flush: 320 contiguous floats per wave, 5 passes of 32 lanes x b64 ----
    float* dst = out + (size_t)g0 * OUT_PER_PAIR;
    #pragma unroll
    for (int pass = 0; pass < 5; ++pass) {
        const v2f v = *(const v2f*)&wls[pass * 64 + lane * 2];
        *(v2f*)&dst[pass * 64 + lane * 2] = v;
    }
}

extern "C" void kernel_launch(void* const* d_in, const int* in_sizes, int n_in,
                              void* d_out, int out_size, void* d_ws, size_t ws_size,
                              hipStream_t stream) {
    const float* x        = (const float*)d_in[0];  // [4096,3,32,32]
    const int*   pair_idx = (const int*)  d_in[1];  // [4096,460,2]
    const float* theta    = (const float*)d_in[2];  // [10]
    float*       out      = (float*)d_out;          // [4096,920,10]

    const int blocks = TOTAL_PAIRS / PAIRS_PER_BLOCK; // 14720, exact cover
    fuzzy_qlayer_wmma<<<blocks, 256, 0, stream>>>(x, pair_idx, theta, out);
}